// _BaseFilter_51178830299244
// MI455X (gfx1250) — compile-verified
//
#include <hip/hip_runtime.h>
#include <hip/hip_bf16.h>

typedef float v2f __attribute__((ext_vector_type(2)));
typedef float v8f __attribute__((ext_vector_type(8)));

#define SEGLEN 1024          // samples per segment (64 blocks of 16)
#define SEGBLK 64            // blocks per segment
#define SEGSHIFT 10

// ---------------------------------------------------------------------------
// K1: per-row max(|x|) reduction.  One block per row.
// ---------------------------------------------------------------------------
__global__ void scale_kernel(const float* __restrict__ x, float* __restrict__ scale, int T) {
    const int r = blockIdx.x;
    const float* xr = x + (size_t)r * T;
    float m = 0.f;
    for (int i = threadIdx.x; i < T; i += blockDim.x) m = fmaxf(m, fabsf(xr[i]));
    __shared__ float red[256];
    red[threadIdx.x] = m;
    __syncthreads();
    for (int s = 128; s > 0; s >>= 1) {
        if ((int)threadIdx.x < s) red[threadIdx.x] = fmaxf(red[threadIdx.x], red[threadIdx.x + s]);
        __syncthreads();
    }
    if (threadIdx.x == 0) scale[r] = red[0];
}

// ---------------------------------------------------------------------------
// gtab_prep: zero-input responses g_m[i], i=0..SEGLEN-1, m=1..4 (one thread per m)
// g_m solves the all-pole recursion with y[-k] = delta_{k,m}.
// ---------------------------------------------------------------------------
__global__ void gtab_prep(const float* __restrict__ ap, float* __restrict__ gtab) {
    const int m = (int)threadIdx.x + 1;
    if (m > 4) return;
    const float a0 = ap[0];
    const float a1 = ap[1] / a0, a2 = ap[2] / a0, a3 = ap[3] / a0, a4 = ap[4] / a0;
    float w1 = (m == 1) ? 1.f : 0.f, w2 = (m == 2) ? 1.f : 0.f;
    float w3 = (m == 3) ? 1.f : 0.f, w4 = (m == 4) ? 1.f : 0.f;
    float* g = gtab + (size_t)(m - 1) * SEGLEN;
    for (int i = 0; i < SEGLEN; ++i) {
        const float yv = -(a1 * w1 + a2 * w2 + a3 * w3 + a4 * w4);
        g[i] = yv;
        w4 = w3; w3 = w2; w2 = w1; w1 = yv;
    }
}

// correction of the zero-state solution at absolute padded index j of row r
__device__ __forceinline__ float seg_corr(const float* __restrict__ gtab,
                                          const float* __restrict__ dbuf,
                                          int r, int j, int nseg) {
    const int s = j >> SEGSHIFT;
    const int i = j & (SEGLEN - 1);
    const float* d = dbuf + ((size_t)r * nseg + s) * 4;
    return gtab[i] * d[0] + gtab[SEGLEN + i] * d[1] +
           gtab[2 * SEGLEN + i] * d[2] + gtab[3 * SEGLEN + i] * d[3];
}

// ---------------------------------------------------------------------------
// K2: normalized odd-reflect padded signal + 5-tap FIR  -> f  (length SP)
// ---------------------------------------------------------------------------
__global__ void fir_fwd(const float* __restrict__ x, const float* __restrict__ scale,
                        const float* __restrict__ bp, const float* __restrict__ ap,
                        float* __restrict__ f, int T, int Tp, int SP) {
    const int r = blockIdx.y;
    const int t = blockIdx.x * blockDim.x + threadIdx.x;
    if (t >= SP) return;
    const float a0 = ap[0];
    const float inv_s = 1.0f / scale[r];
    const float* xr = x + (size_t)r * T;
    float acc = 0.f;
    if (t < Tp) {
#pragma unroll
        for (int k = 0; k < 5; ++k) {
            const int i = t - k;
            if (i < 0) continue;
            float v;
            if (i < 15)          v = 2.f * xr[0]     - xr[14 - i];
            else if (i < T + 15) v = xr[i - 15];
            else                 v = 2.f * xr[T - 1] - xr[2 * T + 12 - i];
            acc += (bp[k] / a0) * (v * inv_s);
        }
    }
    f[(size_t)r * SP + t] = acc;
}

// ---------------------------------------------------------------------------
// K4: FIR on time-reversed forward result, with segment correction fused in.
// ---------------------------------------------------------------------------
__global__ void fir_bwd_corr(const float* __restrict__ y1,
                             const float* __restrict__ bp, const float* __restrict__ ap,
                             const float* __restrict__ gtab, const float* __restrict__ dbuf,
                             float* __restrict__ f2, int Tp, int SP, int nseg) {
    const int r = blockIdx.y;
    const int t = blockIdx.x * blockDim.x + threadIdx.x;
    if (t >= SP) return;
    const float a0 = ap[0];
    const float* yr = y1 + (size_t)r * SP;
    float acc = 0.f;
    if (t < Tp) {
#pragma unroll
        for (int k = 0; k < 5; ++k) {
            const int i = t - k;
            if (i < 0) continue;
            const int j = Tp - 1 - i;
            const float yv = yr[j] + seg_corr(gtab, dbuf, r, j, nseg);
            acc += (bp[k] / a0) * yv;
        }
    }
    f2[(size_t)r * SP + t] = acc;
}

// ---------------------------------------------------------------------------
// K3/K5: zero-state all-pole IIR over ONE segment per wave, via fp32 WMMA.
// blockIdx.x = 16-row tile, blockIdx.y = segment.
// Per 16-sample block:  P = F(16x16)·H^T  (state-independent, 4 WMMAs)
//                       Y = S(16x4)·G^T + P  (1 WMMA, state-dependent tail)
// ---------------------------------------------------------------------------
__global__ __launch_bounds__(32) void iir_wmma_seg(const float* __restrict__ f,
                                                   float* __restrict__ y,
                                                   const float* __restrict__ ap,
                                                   int nblk_total, int SP) {
    const int lane = threadIdx.x;
    const int n  = lane & 15;   // N (time within block) for B/D operands
    const int hi = lane >> 4;
    const int r0 = blockIdx.x * 16;
    const int b0 = blockIdx.y * SEGBLK;
    const int bend = min(b0 + SEGBLK, nblk_total);

    __shared__ float sh_h[16];     // all-pole impulse response h[0..15]
    __shared__ float sh_g[4][16];  // zero-input responses g_m[0..15]
    __shared__ float scol[16][4];  // last 4 output columns (state handoff)

    if (lane == 0) {
        const float a0 = ap[0];
        const float a1 = ap[1] / a0, a2 = ap[2] / a0, a3 = ap[3] / a0, a4 = ap[4] / a0;
        float hh[20];
        hh[0] = hh[1] = hh[2] = hh[3] = 0.f; hh[4] = 1.f;
#pragma unroll
        for (int i = 5; i < 20; ++i)
            hh[i] = -(a1 * hh[i - 1] + a2 * hh[i - 2] + a3 * hh[i - 3] + a4 * hh[i - 4]);
#pragma unroll
        for (int i = 0; i < 16; ++i) sh_h[i] = hh[i + 4];
#pragma unroll
        for (int m = 1; m <= 4; ++m) {
            float gg[20];
            gg[0] = gg[1] = gg[2] = gg[3] = 0.f;
            gg[4 - m] = 1.f;                       // y[t0 - m] = 1
#pragma unroll
            for (int i = 4; i < 20; ++i)
                gg[i] = -(a1 * gg[i - 1] + a2 * gg[i - 2] + a3 * gg[i - 3] + a4 * gg[i - 4]);
#pragma unroll
            for (int i = 0; i < 16; ++i) sh_g[m - 1][i] = gg[i + 4];
        }
    }
    __syncthreads();

    // B operands (constant across the scan).
    // B(4x16) layout: VGPR0 = K (0 | 2), VGPR1 = K (1 | 3), lane&15 = N.
    v2f Bh[4];
#pragma unroll
    for (int c = 0; c < 4; ++c) {
        const int j0 = 4 * c + 2 * hi;             // H^T[j, n] = h[n - j]
        Bh[c].x = (n - j0     >= 0) ? sh_h[n - j0]     : 0.f;
        Bh[c].y = (n - j0 - 1 >= 0) ? sh_h[n - j0 - 1] : 0.f;
    }
    v2f Bg;
    Bg.x = sh_g[2 * hi + 0][n];
    Bg.y = sh_g[2 * hi + 1][n];

    // A(16x4) layout: lane<16 row=lane K={0,1}; lane>=16 row=lane-16 K={2,3}
    const float* frow = f + (size_t)(r0 + n) * SP + 2 * hi;
    float* yrow = y + (size_t)r0 * SP;

    float s0 = 0.f, s1 = 0.f;      // zero state at segment start (fixed up later)

    for (int b = b0; b < bend; ++b) {
        const int t0 = b * 16;
        v2f A0 = *(const v2f*)(frow + t0 + 0);
        v2f A1 = *(const v2f*)(frow + t0 + 4);
        v2f A2 = *(const v2f*)(frow + t0 + 8);
        v2f A3 = *(const v2f*)(frow + t0 + 12);
        __builtin_prefetch(frow + t0 + 64, 0, 1);  // global_prefetch_b8

        // state-independent part first (overlaps the state handoff latency)
        v8f P = {};
        P = __builtin_amdgcn_wmma_f32_16x16x4_f32(false, A0, false, Bh[0], (short)0, P, false, false);
        P = __builtin_amdgcn_wmma_f32_16x16x4_f32(false, A1, false, Bh[1], (short)0, P, false, false);
        P = __builtin_amdgcn_wmma_f32_16x16x4_f32(false, A2, false, Bh[2], (short)0, P, false, false);
        P = __builtin_amdgcn_wmma_f32_16x16x4_f32(false, A3, false, Bh[3], (short)0, P, false, false);

        // single state-dependent WMMA tail
        v2f As; As.x = s0; As.y = s1;
        v8f acc = __builtin_amdgcn_wmma_f32_16x16x4_f32(false, As, false, Bg, (short)0, P, false, false);

        // D layout: VGPR v -> row (v + 8*hi), time n
#pragma unroll
        for (int v = 0; v < 8; ++v)
            yrow[(size_t)(v + 8 * hi) * SP + t0 + n] = acc[v];

        // State handoff: next block needs y[t0+15..t0+12] per row.
        if (n >= 12) {
#pragma unroll
            for (int v = 0; v < 8; ++v)
                scol[v + 8 * hi][n - 12] = acc[v];
        }
        __syncthreads();
        s0 = hi ? scol[n][1] : scol[n][3];         // S[:,2]=y[t0+13] | S[:,0]=y[t0+15]
        s1 = hi ? scol[n][0] : scol[n][2];         // S[:,3]=y[t0+12] | S[:,1]=y[t0+14]
        __syncthreads();
    }
}

// ---------------------------------------------------------------------------
// fixup_scan: propagate true segment-boundary states.  One thread per row.
// delta[s+1]_m = y_zs[t_{s+1}-m] + sum_{m'} g_{m'}[SEGLEN-m] * delta[s]_{m'}
// ---------------------------------------------------------------------------
__global__ void fixup_scan(const float* __restrict__ yzs, const float* __restrict__ gtab,
                           float* __restrict__ dbuf, int SP, int nseg, int R) {
    const int r = blockIdx.x * blockDim.x + threadIdx.x;
    if (r >= R) return;
    const float* yr = yzs + (size_t)r * SP;
    float* drow = dbuf + (size_t)r * nseg * 4;
    float d0 = 0.f, d1 = 0.f, d2 = 0.f, d3 = 0.f;
    drow[0] = drow[1] = drow[2] = drow[3] = 0.f;
    for (int s = 0; s + 1 < nseg; ++s) {
        const int te = SEGLEN * (s + 1);
        float nd[4];
#pragma unroll
        for (int m = 1; m <= 4; ++m) {
            const int i = SEGLEN - m;
            nd[m - 1] = yr[te - m] + gtab[i] * d0 + gtab[SEGLEN + i] * d1 +
                        gtab[2 * SEGLEN + i] * d2 + gtab[3 * SEGLEN + i] * d3;
        }
        d0 = nd[0]; d1 = nd[1]; d2 = nd[2]; d3 = nd[3];
        float* dst = drow + (size_t)(s + 1) * 4;
        dst[0] = d0; dst[1] = d1; dst[2] = d2; dst[3] = d3;
    }
}

// ---------------------------------------------------------------------------
// K6: out[r,t] = (y2_zs[r, off - t] + corr) * scale[r]   (flip + crop padlen)
// ---------------------------------------------------------------------------
__global__ void out_corr(const float* __restrict__ y2, const float* __restrict__ scale,
                         const float* __restrict__ gtab, const float* __restrict__ dbuf,
                         float* __restrict__ out, int T, int SP, int off, int nseg) {
    const int r = blockIdx.y;
    const int t = blockIdx.x * blockDim.x + threadIdx.x;
    if (t >= T) return;
    const int j = off - t;
    const float yv = y2[(size_t)r * SP + j] + seg_corr(gtab, dbuf, r, j, nseg);
    out[(size_t)r * T + t] = yv * scale[r];
}

extern "C" void kernel_launch(void* const* d_in, const int* in_sizes, int n_in,
                              void* d_out, int out_size, void* d_ws, size_t ws_size,
                              hipStream_t stream) {
    const float* x = (const float*)d_in[0];
    const float* b = (const float*)d_in[1];
    const float* a = (const float*)d_in[2];
    float* out = (float*)d_out;

    const int T  = 48000;
    const int R  = in_sizes[0] / T;        // 512 rows (multiple of 16)
    const int Tp = T + 30;                 // padded length 48030
    const int SP = 48032;                  // stride rounded to 16
    const int NBLK = SP / 16;              // 3002 time blocks
    const int NSEG = (NBLK + SEGBLK - 1) / SEGBLK;   // 47 segments

    // workspace layout
    float* scale = (float*)d_ws;                        // 512
    float* gtab  = scale + 512;                         // 4 * SEGLEN
    float* dbuf1 = gtab + 4 * SEGLEN;                   // R * NSEG * 4
    float* dbuf2 = dbuf1 + (size_t)R * NSEG * 4;        // R * NSEG * 4
    float* bufA  = dbuf2 + (size_t)R * NSEG * 4;        // R * SP
    float* bufB  = bufA + (size_t)R * SP;               // R * SP

    scale_kernel<<<R, 256, 0, stream>>>(x, scale, T);
    gtab_prep<<<1, 4, 0, stream>>>(a, gtab);

    dim3 gP((SP + 255) / 256, R);
    fir_fwd<<<gP, 256, 0, stream>>>(x, scale, b, a, bufA, T, Tp, SP);

    dim3 gI(R / 16, NSEG);
    iir_wmma_seg<<<gI, 32, 0, stream>>>(bufA, bufB, a, NBLK, SP);     // forward, zero-state
    fixup_scan<<<(R + 255) / 256, 256, 0, stream>>>(bufB, gtab, dbuf1, SP, NSEG, R);

    fir_bwd_corr<<<gP, 256, 0, stream>>>(bufB, b, a, gtab, dbuf1, bufA, Tp, SP, NSEG);

    iir_wmma_seg<<<gI, 32, 0, stream>>>(bufA, bufB, a, NBLK, SP);     // backward, zero-state
    fixup_scan<<<(R + 255) / 256, 256, 0, stream>>>(bufB, gtab, dbuf2, SP, NSEG, R);

    dim3 gO((T + 255) / 256, R);
    out_corr<<<gO, 256, 0, stream>>>(bufB, scale, gtab, dbuf2, out, T, SP, Tp - 16, NSEG);
}